// KronEmbedding_14465449853576
// MI455X (gfx1250) — compile-verified
//
#include <hip/hip_runtime.h>

typedef __attribute__((ext_vector_type(2))) float v2f;
typedef __attribute__((ext_vector_type(8))) float v8f;

// Problem constants (match reference): R=8, M1=224, N1=64, M2=224, N2=32
#define R_  8
#define M1_ 224
#define N1_ 64
#define M2_ 224
#define N2_ 32
// VOCAB = 224*224 = 50176, EMB = 64*32 = 2048

// One wave32 per token. Out(64x32) = A_i^T(64x8) x B_j(8x32) computed as
// 4x2 tiles of 16x16, each via two chained V_WMMA_F32_16X16X4_F32 (K=4+4).
__global__ __launch_bounds__(256) void kron_embed_wmma(
    const float* __restrict__ A,   // [R, M1, N1]
    const float* __restrict__ B,   // [R, M2, N2]
    const int*   __restrict__ x,   // [n_tokens]
    float*       __restrict__ out, // [n_tokens, EMB]
    int n_tokens)
{
    const int wave_in_block = threadIdx.x >> 5;
    const int t = blockIdx.x * (blockDim.x >> 5) + wave_in_block;
    if (t >= n_tokens) return;           // whole wave exits -> EXEC all-1 at WMMA

    const int lane = threadIdx.x & 31;
    const int half = lane >> 4;          // 0: lanes 0-15, 1: lanes 16-31
    const int mn   = lane & 15;          // M (for A) / N (for B,C,D)

    const int v = x[t];
    const int i = v / M2_;
    const int j = v - i * M2_;

    const float* Ai = A + (size_t)i * N1_;   // A[s,i,a] = Ai[s*M1_*N1_ + a]
    const float* Bj = B + (size_t)j * N2_;   // B[s,j,b] = Bj[s*M2_*N2_ + b]

    // A-operand tiles: Amat[m,k] = A[k, i, 16*ta + m]
    // f32 16x4 layout: lane holds M=mn; VGPR0 = K=2*half, VGPR1 = K=2*half+1
    v2f amat[4][2];
#pragma unroll
    for (int ta = 0; ta < 4; ++ta) {
#pragma unroll
        for (int kk = 0; kk < 2; ++kk) {            // kk=0 -> K 0..3, kk=1 -> K 4..7
            const int k0 = kk * 4 + 2 * half;
            amat[ta][kk].x = Ai[(size_t)(k0    ) * (M1_ * N1_) + 16 * ta + mn];
            amat[ta][kk].y = Ai[(size_t)(k0 + 1) * (M1_ * N1_) + 16 * ta + mn];
        }
    }

    // B-operand tiles: Bmat[k,n] = B[k, j, 16*tb + n] (mirrored layout: lane = N)
    v2f bmat[2][2];
#pragma unroll
    for (int tb = 0; tb < 2; ++tb) {
#pragma unroll
        for (int kk = 0; kk < 2; ++kk) {
            const int k0 = kk * 4 + 2 * half;
            bmat[tb][kk].x = Bj[(size_t)(k0    ) * (M2_ * N2_) + 16 * tb + mn];
            bmat[tb][kk].y = Bj[(size_t)(k0 + 1) * (M2_ * N2_) + 16 * tb + mn];
        }
    }

    // 8 accumulator tiles (64 VGPRs), K=8 via two chained K=4 WMMAs.
    v8f acc[4][2];
#pragma unroll
    for (int ta = 0; ta < 4; ++ta) {
#pragma unroll
        for (int tb = 0; tb < 2; ++tb) {
            v8f c = {0.f, 0.f, 0.f, 0.f, 0.f, 0.f, 0.f, 0.f};
            // 8 args: (neg_a, A, neg_b, B, c_mod, C, reuse_a, reuse_b)
            c = __builtin_amdgcn_wmma_f32_16x16x4_f32(
                    false, amat[ta][0], false, bmat[tb][0], (short)0, c, false, false);
            c = __builtin_amdgcn_wmma_f32_16x16x4_f32(
                    false, amat[ta][1], false, bmat[tb][1], (short)0, c, false, false);
            acc[ta][tb] = c;
        }
    }

    // Stream results out with non-temporal stores (64 MB written once, never
    // re-read; keep L2 free for the 687 KB A/B working set).
    // C/D layout: lane -> N = mn, VGPR v -> M = v + 8*half.
    float* outT = out + (size_t)t * (N1_ * N2_);
#pragma unroll
    for (int ta = 0; ta < 4; ++ta) {
#pragma unroll
        for (int tb = 0; tb < 2; ++tb) {
#pragma unroll
            for (int vv = 0; vv < 8; ++vv) {
                const int row = 16 * ta + vv + 8 * half;   // a in [0,64)
                const int col = 16 * tb + mn;              // b in [0,32)
                __builtin_nontemporal_store(acc[ta][tb][vv],
                                            &outT[row * N2_ + col]);
            }
        }
    }
}

extern "C" void kernel_launch(void* const* d_in, const int* in_sizes, int n_in,
                              void* d_out, int out_size, void* d_ws, size_t ws_size,
                              hipStream_t stream) {
    const float* A = (const float*)d_in[0];
    const float* B = (const float*)d_in[1];
    const int*   x = (const int*)d_in[2];
    float*     out = (float*)d_out;

    const int n_tokens = in_sizes[2];            // 4*2048 = 8192
    const int block = 256;                        // 8 waves per block
    const int waves_per_block = block / 32;
    const int grid = (n_tokens + waves_per_block - 1) / waves_per_block;

    kron_embed_wmma<<<grid, block, 0, stream>>>(A, B, x, out, n_tokens);
}